// SAGEConv_88244398064425
// MI455X (gfx1250) — compile-verified
//
#include <hip/hip_runtime.h>
#include <hip/hip_bf16.h>

typedef float v2f __attribute__((ext_vector_type(2)));
typedef float v8f __attribute__((ext_vector_type(8)));

#define DFEAT 128
#define TILE_M 16
#define LDS_STRIDE 132   // 128 + 4 pad -> conflict-free A-fragment loads

__global__ __launch_bounds__(128) void sage_fused_wmma(
    const float* __restrict__ x,   const int* __restrict__ row,
    const int* __restrict__ col,   const float* __restrict__ val,
    const float* __restrict__ Wl,  const float* __restrict__ bl,
    const float* __restrict__ Wr,  float* __restrict__ out,
    int n_nodes, int n_edges)
{
    __shared__ float lds_a[TILE_M * LDS_STRIDE];  // aggregated messages tile
    __shared__ float lds_x[TILE_M * LDS_STRIDE];  // raw x tile

    const int t  = threadIdx.x;            // 0..127, thread t owns feature col t
    const int m0 = blockIdx.x * TILE_M;

    // ---- zero my exclusive column of the aggregation tile ----
    #pragma unroll
    for (int r = 0; r < TILE_M; ++r)
        lds_a[r * LDS_STRIDE + t] = 0.0f;

    // ---- stage x tile into LDS (vectorized b128 loads, clamped for tails) ----
    {
        const float4* xv = reinterpret_cast<const float4*>(x);
        #pragma unroll
        for (int i = 0; i < 4; ++i) {
            int j    = t + i * 128;        // 0..511 float4 slots in the tile
            int r    = j >> 5;             // tile row 0..15
            int c4   = j & 31;             // float4 column 0..31
            int gr   = m0 + r; if (gr > n_nodes - 1) gr = n_nodes - 1;
            float4 v4 = xv[(size_t)gr * (DFEAT / 4) + c4];
            *reinterpret_cast<float4*>(&lds_x[r * LDS_STRIDE + c4 * 4]) = v4;
        }
    }

    // ---- edge range for this 16-node tile (row[] is sorted; uniform search) ----
    int e0, e1;
    {
        int lo = 0, hi = n_edges;
        while (lo < hi) { int mid = (lo + hi) >> 1; if (row[mid] < m0) lo = mid + 1; else hi = mid; }
        e0 = lo;
        hi = n_edges;
        while (lo < hi) { int mid = (lo + hi) >> 1; if (row[mid] < m0 + TILE_M) lo = mid + 1; else hi = mid; }
        e1 = lo;
    }

    // ---- SpMM: serialized edge walk, register run-accumulation per row ----
    // thread t touches only column t of lds_a -> race-free, no atomics.
    float acc = 0.0f;
    int   cur = -1;
    for (int e = e0; e < e1; ++e) {
        int   r = row[e];                  // uniform (scalarizable) loads
        int   c = col[e];
        float v = val[e];
        if (r != cur) {
            if (cur >= 0) lds_a[(cur - m0) * LDS_STRIDE + t] = acc;
            acc = 0.0f;
            cur = r;
        }
        acc = fmaf(v, x[(size_t)c * DFEAT + t], acc);   // coalesced 512B gather/edge
    }
    if (cur >= 0) lds_a[(cur - m0) * LDS_STRIDE + t] = acc;

    __syncthreads();

    // ---- dense part on WMMA pipes: out_tile = A_agg@Wl + A_x@Wr + bl ----
    const int wave = t >> 5;               // 0..3 ; each wave owns 2 N-tiles
    const int lane = t & 31;
    const int m    = lane & 15;            // A-fragment row
    const int kk   = (lane >> 4) << 1;     // K sub-offset: 0 (lanes 0-15) / 2 (16-31)
    const int n0   = wave * 32;            // this wave's first N-tile base
    const int nl   = lane & 15;            // B/C column within tile

    const float* __restrict__ lAa = &lds_a[m * LDS_STRIDE + kk];
    const float* __restrict__ lAx = &lds_x[m * LDS_STRIDE + kk];

    v8f c0 = {}; v8f c1 = {};
    for (int k0 = 0; k0 < DFEAT; k0 += 4) {
        // A fragments (f32 16x16x4 layout): VGPR0=K(kk), VGPR1=K(kk+1)
        v2f aL = { lAa[k0], lAa[k0 + 1] };
        v2f aX = { lAx[k0], lAx[k0 + 1] };
        // B fragments from global weights (hot in L2 across all blocks)
        const float* wl = &Wl[(size_t)(k0 + kk) * DFEAT + nl];
        const float* wr = &Wr[(size_t)(k0 + kk) * DFEAT + nl];
        v2f bL0 = { wl[n0],      wl[n0 + DFEAT] };
        v2f bL1 = { wl[n0 + 16], wl[n0 + 16 + DFEAT] };
        v2f bR0 = { wr[n0],      wr[n0 + DFEAT] };
        v2f bR1 = { wr[n0 + 16], wr[n0 + 16 + DFEAT] };
        // interleave the two accumulators -> distance-2 between dependent WMMAs
        c0 = __builtin_amdgcn_wmma_f32_16x16x4_f32(false, aL, false, bL0, (short)0, c0, false, false);
        c1 = __builtin_amdgcn_wmma_f32_16x16x4_f32(false, aL, false, bL1, (short)0, c1, false, false);
        c0 = __builtin_amdgcn_wmma_f32_16x16x4_f32(false, aX, false, bR0, (short)0, c0, false, false);
        c1 = __builtin_amdgcn_wmma_f32_16x16x4_f32(false, aX, false, bR1, (short)0, c1, false, false);
    }

    // ---- epilogue: bias + store (C layout: VGPR j -> M = j + 8*(lane>=16)) ----
    const int mhi  = (lane >> 4) * 8;
    const float b0 = bl[n0 + nl];
    const float b1 = bl[n0 + 16 + nl];
    #pragma unroll
    for (int j = 0; j < 8; ++j) {
        int mr = m0 + mhi + j;
        if (mr < n_nodes) {
            out[(size_t)mr * DFEAT + n0 + nl]      = c0[j] + b0;
            out[(size_t)mr * DFEAT + n0 + 16 + nl] = c1[j] + b1;
        }
    }
}

extern "C" void kernel_launch(void* const* d_in, const int* in_sizes, int n_in,
                              void* d_out, int out_size, void* d_ws, size_t ws_size,
                              hipStream_t stream) {
    const float* x   = (const float*)d_in[0];
    const int*   row = (const int*)  d_in[1];
    const int*   col = (const int*)  d_in[2];
    const float* val = (const float*)d_in[3];
    const float* Wl  = (const float*)d_in[4];
    const float* bl  = (const float*)d_in[5];
    const float* Wr  = (const float*)d_in[6];
    float*       out = (float*)d_out;

    const int n_nodes = in_sizes[0] / DFEAT;
    const int n_edges = in_sizes[1];
    const int grid    = (n_nodes + TILE_M - 1) / TILE_M;   // 6250 for N=100000

    sage_fused_wmma<<<grid, 128, 0, stream>>>(x, row, col, val, Wl, bl, Wr, out,
                                              n_nodes, n_edges);
}